// AttFlowLayer_1580547965298
// MI455X (gfx1250) — compile-verified
//
#include <hip/hip_runtime.h>
#include <hip/hip_bf16.h>
#include <math.h>

typedef __attribute__((ext_vector_type(2))) float v2f;
typedef __attribute__((ext_vector_type(8))) float v8f;

#define B_  8
#define LC_ 2048
#define LQ_ 1024
#define D_  256

// ---------------------------------------------------------------------------
// Kernel 0: row dots cdot = C.w_c, qdot = Q.w_q, qmask = sign|sum Q|; zero colsum
// One wave32 per row. 16384 context rows + 1024 query rows = 17408 waves.
// ---------------------------------------------------------------------------
__global__ void prep_kernel(const float* __restrict__ C, const float* __restrict__ Q,
                            const float* __restrict__ w, float* __restrict__ cdot,
                            float* __restrict__ qdot, float* __restrict__ qmask,
                            float* __restrict__ colsum) {
    const int wid  = blockIdx.x * 8 + (threadIdx.x >> 5);
    const int lane = threadIdx.x & 31;
    if (wid < B_ * LC_) {
        const float* row = C + (size_t)wid * D_;
        const float* wc  = w;                       // w_alpha[0:D]
        float s = 0.f;
        #pragma unroll
        for (int k = lane; k < D_; k += 32) s += row[k] * wc[k];
        #pragma unroll
        for (int off = 16; off; off >>= 1) s += __shfl_xor(s, off, 32);
        if (lane == 0) { cdot[wid] = s; colsum[wid] = 0.0f; }
    } else {
        const int j = wid - B_ * LC_;               // < LQ
        const float* row = Q + (size_t)j * D_;
        const float* wq  = w + D_;                  // w_alpha[D:2D]
        float s = 0.f, r = 0.f;
        #pragma unroll
        for (int k = lane; k < D_; k += 32) { float q = row[k]; s += q * wq[k]; r += q; }
        #pragma unroll
        for (int off = 16; off; off >>= 1) {
            s += __shfl_xor(s, off, 32);
            r += __shfl_xor(r, off, 32);
        }
        if (lane == 0) { qdot[j] = s; qmask[j] = (r != 0.0f) ? 1.0f : 0.0f; }
    }
}

// ---------------------------------------------------------------------------
// Kernel 1: ST[b][j][i] = (C[b,i] * w_m) . Q[j]   (biases added in softmax pass)
// Per-batch GEMM: Mdim=j (LQ), Ndim=i (LC), K=D, via V_WMMA_F32_16X16X4_F32.
// Block = 256 thr = 8 waves laid out 2(j) x 4(i); each wave does a 16x16 tile.
// A-matrix lane layout (16x4 f32): lane%16 = M, half = lane/16 picks K pair.
// ---------------------------------------------------------------------------
__global__ void s_gemm_kernel(const float* __restrict__ C, const float* __restrict__ Q,
                              const float* __restrict__ w, float* __restrict__ ST) {
    const int lane = threadIdx.x & 31;
    const int wave = threadIdx.x >> 5;
    const int wj   = wave & 1;
    const int wi   = wave >> 1;
    const int b    = blockIdx.z;
    const int j0   = blockIdx.x * 32 + wj * 16;
    const int i0   = blockIdx.y * 64 + wi * 16;
    const int half = lane >> 4;
    const int lr   = lane & 15;

    const float* __restrict__ wm   = w + 2 * D_;                         // w_alpha[2D:3D]
    const float* __restrict__ qrow = Q + (size_t)(j0 + lr) * D_;          // A row (M=j)
    const float* __restrict__ crow = C + ((size_t)b * LC_ + i0 + lr) * D_; // B col (N=i)

    __builtin_prefetch(qrow, 0, 1);   // global_prefetch_b8
    __builtin_prefetch(crow, 0, 1);

    v8f acc = {};
    #pragma unroll 8
    for (int k = 0; k < D_; k += 4) {
        const int kk = k + half * 2;
        v2f a, bm;
        a.x  = qrow[kk];
        a.y  = qrow[kk + 1];
        bm.x = crow[kk]     * wm[kk];
        bm.y = crow[kk + 1] * wm[kk + 1];
        acc = __builtin_amdgcn_wmma_f32_16x16x4_f32(false, a, false, bm,
                                                    (short)0, acc, false, false);
    }

    float* __restrict__ out = ST + ((size_t)b * LQ_ + j0) * LC_ + i0;
    #pragma unroll
    for (int r = 0; r < 8; ++r) {
        const int row = r + half * 8;            // M (j offset); N (i offset) = lr
        out[(size_t)row * LC_ + lr] = acc[r];
    }
}

// ---------------------------------------------------------------------------
// Kernel 2: softmax over i for each (b,j) row of ST; adds biases, applies
// qmask, writes normalized A in place, atomically accumulates colsum[b,i].
// One 256-thread block per (b,j) row; 8 elements per thread.
// ---------------------------------------------------------------------------
__global__ void softmax_kernel(float* __restrict__ ST, const float* __restrict__ cdot,
                               const float* __restrict__ qdot, const float* __restrict__ qmask,
                               float* __restrict__ colsum) {
    const int bj = blockIdx.x;                   // 0 .. B*LQ-1
    const int b  = bj >> 10;                     // LQ_ == 1024
    const int j  = bj & (LQ_ - 1);
    float* __restrict__ row = ST + (size_t)bj * LC_;
    const float* __restrict__ cd = cdot + b * LC_;
    const float qd = qdot[j];
    const float qm = qmask[j];
    const int t = threadIdx.x;

    float v[8];
    float mx = -INFINITY;
    #pragma unroll
    for (int u = 0; u < 8; ++u) {
        const int i = t + u * 256;
        v[u] = row[i] + cd[i] + qd;
        mx = fmaxf(mx, v[u]);
    }

    __shared__ float red[256];
    red[t] = mx;
    __syncthreads();
    #pragma unroll
    for (int s = 128; s > 0; s >>= 1) {
        if (t < s) red[t] = fmaxf(red[t], red[t + s]);
        __syncthreads();
    }
    mx = red[0];
    __syncthreads();

    float sum = 0.f;
    #pragma unroll
    for (int u = 0; u < 8; ++u) {
        const float p = __expf(v[u] - mx);
        v[u] = p;
        sum += p;
    }
    red[t] = sum;
    __syncthreads();
    #pragma unroll
    for (int s = 128; s > 0; s >>= 1) {
        if (t < s) red[t] += red[t + s];
        __syncthreads();
    }
    sum = red[0];

    const float inv = qm / sum;
    #pragma unroll
    for (int u = 0; u < 8; ++u) {
        const int i = t + u * 256;
        const float a = v[u] * inv;
        row[i] = a;
        atomicAdd(&colsum[b * LC_ + i], a);      // global_atomic_add_f32
    }
}

// ---------------------------------------------------------------------------
// Kernel 3: H[b,j,d] = sum_i A[b,j,i] * C[b,i,d]
// Per-batch GEMM: M=LQ, N=D, K=LC, f32 WMMA 16x16x4.
// Block = 256 thr = 8 waves laid out 4(j) x 2(d).
// ---------------------------------------------------------------------------
__global__ void h_gemm_kernel(const float* __restrict__ A, const float* __restrict__ C,
                              float* __restrict__ H) {
    const int lane = threadIdx.x & 31;
    const int wave = threadIdx.x >> 5;
    const int wd   = wave & 1;
    const int wj   = wave >> 1;
    const int b    = blockIdx.z;
    const int d0   = blockIdx.x * 32 + wd * 16;
    const int j0   = blockIdx.y * 64 + wj * 16;
    const int half = lane >> 4;
    const int lr   = lane & 15;

    const float* __restrict__ arow = A + ((size_t)b * LQ_ + j0 + lr) * LC_;  // A row (M=j)
    const float* __restrict__ bcol = C + (size_t)b * LC_ * D_ + d0 + lr;     // B col (N=d)

    __builtin_prefetch(arow, 0, 1);
    __builtin_prefetch(bcol, 0, 1);

    v8f acc = {};
    #pragma unroll 4
    for (int k = 0; k < LC_; k += 4) {
        const int kk = k + half * 2;
        v2f a, bb;
        a.x  = arow[kk];
        a.y  = arow[kk + 1];
        bb.x = bcol[(size_t)kk * D_];
        bb.y = bcol[(size_t)(kk + 1) * D_];
        acc = __builtin_amdgcn_wmma_f32_16x16x4_f32(false, a, false, bb,
                                                    (short)0, acc, false, false);
    }

    float* __restrict__ out = H + ((size_t)b * LQ_ + j0) * D_ + d0;
    #pragma unroll
    for (int r = 0; r < 8; ++r) {
        const int row = r + half * 8;            // M (j offset); N (d offset) = lr
        out[(size_t)row * D_ + lr] = acc[r];
    }
}

// ---------------------------------------------------------------------------
// Kernel 4: G[b,i,0:D] = C[b,i,:] ; G[b,i,D:2D] = C[b,i,:] * colsum[b,i]
// ---------------------------------------------------------------------------
__global__ void g_out_kernel(const float* __restrict__ C, const float* __restrict__ colsum,
                             float* __restrict__ G) {
    const size_t idx = (size_t)blockIdx.x * 256 + threadIdx.x;   // over B*LC*D
    const int d  = (int)(idx & (D_ - 1));
    const size_t bi = idx >> 8;                                  // b*LC + i
    const float c  = C[idx];
    const float cs = colsum[bi];
    float* __restrict__ grow = G + bi * (2 * D_);
    grow[d]       = c;
    grow[D_ + d]  = c * cs;
}

// ---------------------------------------------------------------------------
extern "C" void kernel_launch(void* const* d_in, const int* in_sizes, int n_in,
                              void* d_out, int out_size, void* d_ws, size_t ws_size,
                              hipStream_t stream) {
    const float* C = (const float*)d_in[0];   // (B, LC, D)
    const float* Q = (const float*)d_in[1];   // (LQ, D)
    const float* W = (const float*)d_in[2];   // (3D,)

    float* G = (float*)d_out;                                   // B*LC*2D
    float* H = (float*)d_out + (size_t)B_ * LC_ * 2 * D_;       // B*LQ*D

    // Workspace layout
    char*  ws     = (char*)d_ws;
    float* ST     = (float*)ws;                                        // 64 MB: A / S^T [b][j][i]
    size_t off    = (size_t)B_ * LQ_ * LC_ * sizeof(float);
    float* cdot   = (float*)(ws + off); off += (size_t)B_ * LC_ * sizeof(float);
    float* qdot   = (float*)(ws + off); off += (size_t)LQ_ * sizeof(float);
    float* qmask  = (float*)(ws + off); off += (size_t)LQ_ * sizeof(float);
    float* colsum = (float*)(ws + off);

    // 0) row dots + masks + zero colsum: (16384 + 1024) waves, 8 waves/block
    prep_kernel<<<(B_ * LC_ + LQ_) / 8, 256, 0, stream>>>(C, Q, W, cdot, qdot, qmask, colsum);

    // 1) S^T GEMM: grid (LQ/32, LC/64, B)
    s_gemm_kernel<<<dim3(LQ_ / 32, LC_ / 64, B_), 256, 0, stream>>>(C, Q, W, ST);

    // 2) softmax rows + colsum atomics: one block per (b,j)
    softmax_kernel<<<B_ * LQ_, 256, 0, stream>>>(ST, cdot, qdot, qmask, colsum);

    // 3) H GEMM: grid (D/32, LQ/64, B)
    h_gemm_kernel<<<dim3(D_ / 32, LQ_ / 64, B_), 256, 0, stream>>>(ST, C, H);

    // 4) G concat: B*LC*D / 256 blocks
    g_out_kernel<<<(B_ * LC_ * D_) / 256, 256, 0, stream>>>(C, colsum, G);
}